// RankGCN_3461743640870
// MI455X (gfx1250) — compile-verified
//
#include <hip/hip_runtime.h>
#include <hip/hip_bf16.h>

#define N_NODES 100000
#define N_EDGES 3200000
#define HID 64
#define RED 32
#define EMB 32
#define LN_EPS 1e-5f

typedef __attribute__((ext_vector_type(2))) float v2f;
typedef __attribute__((ext_vector_type(8))) float v8f;

// ---------------- degree ----------------
__global__ void k_deg_init(float* __restrict__ deg) {
    int i = blockIdx.x * blockDim.x + threadIdx.x;
    if (i < N_NODES) deg[i] = 1.0f;  // self-loop weight
}

__global__ void k_deg_scatter(const int* __restrict__ col, const float* __restrict__ w,
                              float* __restrict__ deg) {
    int e = blockIdx.x * blockDim.x + threadIdx.x;
    if (e < N_EDGES) atomicAdd(&deg[col[e]], w[e]);
}

// ------------- dinv + layer1 init: acc1 = dinv^2*(x@W1) + b1 + x@rW1 + rb1 -------------
__global__ void k_l1_init(const float* __restrict__ x, const float* __restrict__ deg,
                          const float* __restrict__ W1, const float* __restrict__ b1,
                          const float* __restrict__ rW1, const float* __restrict__ rb1,
                          float* __restrict__ dinv, float* __restrict__ xw1,
                          float* __restrict__ acc1) {
    long long t = (long long)blockIdx.x * blockDim.x + threadIdx.x;
    if (t >= (long long)N_NODES * HID) return;
    int n = (int)(t >> 6);
    int f = (int)(t & 63);
    float d  = deg[n];
    float di = d > 0.f ? rsqrtf(d) : 0.f;
    if (f == 0) dinv[n] = di;
    float x0 = x[2 * n], x1v = x[2 * n + 1];
    float xw = fmaf(x0, W1[f], x1v * W1[HID + f]);
    xw1[t]  = xw;
    acc1[t] = di * di * xw + b1[f] + fmaf(x0, rW1[f], x1v * rW1[HID + f]) + rb1[f];
}

// ------------- edge scatter: acc[col] += dinv[row]*w*dinv[col] * src[row] -------------
template <int F>
__global__ void k_edge_scatter(const int* __restrict__ row, const int* __restrict__ col,
                               const float* __restrict__ w, const float* __restrict__ dinv,
                               const float* __restrict__ src, float* __restrict__ acc) {
    long long t = (long long)blockIdx.x * blockDim.x + threadIdx.x;
    if (t >= (long long)N_EDGES * F) return;
    int e = (int)(t / F);
    int f = (int)(t & (F - 1));
    int r = row[e], c = col[e];
    float coeff = dinv[r] * w[e] * dinv[c];
    atomicAdd(&acc[(long long)c * F + f], coeff * src[(long long)r * F + f]);
}

// ------------- LayerNorm + ReLU over 64 feats (1 wave/node, 2 elems/lane) -------------
__global__ void k_ln_relu_64(const float* __restrict__ acc, const float* __restrict__ g,
                             const float* __restrict__ be, float* __restrict__ out) {
    int wave = threadIdx.x >> 5;
    int lane = threadIdx.x & 31;
    int n = blockIdx.x * (blockDim.x >> 5) + wave;
    if (n >= N_NODES) return;
    float v0 = acc[(long long)n * 64 + lane];
    float v1 = acc[(long long)n * 64 + lane + 32];
    float s = v0 + v1, sq = fmaf(v0, v0, v1 * v1);
    #pragma unroll
    for (int off = 16; off > 0; off >>= 1) {
        s  += __shfl_xor(s,  off, 32);
        sq += __shfl_xor(sq, off, 32);
    }
    float mu  = s * (1.f / 64.f);
    float var = sq * (1.f / 64.f) - mu * mu;
    float inv = rsqrtf(var + LN_EPS);
    float o0 = (v0 - mu) * inv * g[lane]      + be[lane];
    float o1 = (v1 - mu) * inv * g[lane + 32] + be[lane + 32];
    out[(long long)n * 64 + lane]      = fmaxf(o0, 0.f);
    out[(long long)n * 64 + lane + 32] = fmaxf(o1, 0.f);
}

// ------------- Layer2 dense: xw2 = x1@W2, acc2 = dinv^2*xw2 + b2 + x1@rW2 + rb2 -------------
// One wave per 16x16 output tile, K=64 in 16 steps of V_WMMA_F32_16X16X4_F32,
// two accumulators (W2 and rW2) sharing each A fragment.
__global__ void k_l2_wmma(const float* __restrict__ x1, const float* __restrict__ W2,
                          const float* __restrict__ rW2, const float* __restrict__ b2,
                          const float* __restrict__ rb2, const float* __restrict__ dinv,
                          float* __restrict__ xw2, float* __restrict__ acc2) {
    int wave_global = (blockIdx.x * blockDim.x + threadIdx.x) >> 5;
    int lane = threadIdx.x & 31;
    const int tiles_m = N_NODES / 16;           // 6250
    int tm = wave_global >> 1;                  // node tile
    int tn = wave_global & 1;                   // output-col tile (RED/16 == 2)
    if (tm >= tiles_m) return;                  // whole-wave uniform -> EXEC all-1s
    int m0 = tm * 16, n0 = tn * 16;
    int  lr = lane & 15;
    bool hi = lane >= 16;
    int  kofs = hi ? 2 : 0;

    v8f c0 = {};  // x1 @ W2
    v8f c1 = {};  // x1 @ rW2
    #pragma unroll
    for (int k0 = 0; k0 < HID; k0 += 4) {
        // A fragment: 16x4 fp32; lanes 0-15 hold K=k0,k0+1; lanes 16-31 hold K=k0+2,k0+3
        const float* ap = &x1[(long long)(m0 + lr) * HID + k0 + kofs];
        v2f a; a.x = ap[0]; a.y = ap[1];
        // B fragments: 4x16 fp32, row-major weights W[k][n]
        int kb = k0 + kofs;
        v2f b;  b.x  = W2 [kb * RED + n0 + lr]; b.y  = W2 [(kb + 1) * RED + n0 + lr];
        v2f br; br.x = rW2[kb * RED + n0 + lr]; br.y = rW2[(kb + 1) * RED + n0 + lr];
        c0 = __builtin_amdgcn_wmma_f32_16x16x4_f32(false, a, false, b,  (short)0, c0, false, false);
        c1 = __builtin_amdgcn_wmma_f32_16x16x4_f32(false, a, false, br, (short)0, c1, false, false);
    }
    // D layout: VGPR v -> row m0+v (lanes 0-15) / m0+v+8 (lanes 16-31); col = n0 + lane%16
    int ncol = n0 + lr;
    float b2v = b2[ncol], rb2v = rb2[ncol];
    #pragma unroll
    for (int v = 0; v < 8; ++v) {
        int m = m0 + v + (hi ? 8 : 0);
        float di = dinv[m];
        float xv = c0[v];
        xw2 [(long long)m * RED + ncol] = xv;
        acc2[(long long)m * RED + ncol] = di * di * xv + b2v + c1[v] + rb2v;
    }
}

// ------------- LN2 + ReLU + heads (1 wave/node, 1 feat/lane) -------------
__global__ void k_head(const float* __restrict__ acc2, const float* __restrict__ g2,
                       const float* __restrict__ be2, const float* __restrict__ fcW,
                       const float* __restrict__ fcb, const float* __restrict__ pW,
                       const float* __restrict__ pb, float* __restrict__ logits,
                       float* __restrict__ emb) {
    int wave = threadIdx.x >> 5;
    int lane = threadIdx.x & 31;
    int n = blockIdx.x * (blockDim.x >> 5) + wave;
    if (n >= N_NODES) return;
    float v = acc2[(long long)n * 32 + lane];
    float s = v, sq = v * v;
    #pragma unroll
    for (int off = 16; off > 0; off >>= 1) {
        s  += __shfl_xor(s,  off, 32);
        sq += __shfl_xor(sq, off, 32);
    }
    float mu  = s * (1.f / 32.f);
    float var = sq * (1.f / 32.f) - mu * mu;
    float inv = rsqrtf(var + LN_EPS);
    float x2 = fmaxf((v - mu) * inv * g2[lane] + be2[lane], 0.f);

    // logits = x2 . fcW + fcb
    float lg = x2 * fcW[lane];
    #pragma unroll
    for (int off = 16; off > 0; off >>= 1) lg += __shfl_xor(lg, off, 32);
    if (lane == 0) logits[n] = lg + fcb[0];

    // proj[lane] = sum_i x2[i] * pW[i][lane] + pb[lane]; then L2 normalize
    float p = pb[lane];
    #pragma unroll
    for (int i = 0; i < 32; ++i)
        p = fmaf(__shfl(x2, i, 32), pW[i * 32 + lane], p);
    float nsq = p * p;
    #pragma unroll
    for (int off = 16; off > 0; off >>= 1) nsq += __shfl_xor(nsq, off, 32);
    float nrm = sqrtf(nsq);
    emb[(long long)n * 32 + lane] = p / fmaxf(nrm, 1e-12f);
}

extern "C" void kernel_launch(void* const* d_in, const int* in_sizes, int n_in,
                              void* d_out, int out_size, void* d_ws, size_t ws_size,
                              hipStream_t stream) {
    const float* x    = (const float*)d_in[0];
    const float* ew   = (const float*)d_in[1];
    const float* W1   = (const float*)d_in[2];
    const float* b1   = (const float*)d_in[3];
    const float* rW1  = (const float*)d_in[4];
    const float* rb1  = (const float*)d_in[5];
    const float* g1   = (const float*)d_in[6];
    const float* be1  = (const float*)d_in[7];
    const float* W2   = (const float*)d_in[8];
    const float* b2   = (const float*)d_in[9];
    const float* rW2  = (const float*)d_in[10];
    const float* rb2  = (const float*)d_in[11];
    const float* g2   = (const float*)d_in[12];
    const float* be2  = (const float*)d_in[13];
    const float* fcW  = (const float*)d_in[14];
    const float* fcb  = (const float*)d_in[15];
    const float* pW   = (const float*)d_in[16];
    const float* pb   = (const float*)d_in[17];
    const int*   ei   = (const int*)d_in[18];
    const int*   row  = ei;
    const int*   col  = ei + N_EDGES;

    float* ws   = (float*)d_ws;
    float* deg  = ws;  ws += N_NODES;
    float* dinv = ws;  ws += N_NODES;
    float* xw1  = ws;  ws += (size_t)N_NODES * HID;
    float* acc1 = ws;  ws += (size_t)N_NODES * HID;
    float* x1   = ws;  ws += (size_t)N_NODES * HID;
    float* xw2  = ws;  ws += (size_t)N_NODES * RED;
    float* acc2 = ws;  ws += (size_t)N_NODES * RED;

    float* logits = (float*)d_out;
    float* emb    = logits + N_NODES;

    const int B = 256;
    k_deg_init<<<(N_NODES + B - 1) / B, B, 0, stream>>>(deg);
    k_deg_scatter<<<(N_EDGES + B - 1) / B, B, 0, stream>>>(col, ew, deg);

    long long t1 = (long long)N_NODES * HID;
    k_l1_init<<<(unsigned)((t1 + B - 1) / B), B, 0, stream>>>(x, deg, W1, b1, rW1, rb1,
                                                              dinv, xw1, acc1);

    long long te1 = (long long)N_EDGES * HID;
    k_edge_scatter<HID><<<(unsigned)((te1 + B - 1) / B), B, 0, stream>>>(row, col, ew, dinv,
                                                                          xw1, acc1);

    k_ln_relu_64<<<(N_NODES + 7) / 8, B, 0, stream>>>(acc1, g1, be1, x1);

    int waves_l2 = (N_NODES / 16) * (RED / 16);     // 12500 waves
    int thr_l2   = waves_l2 * 32;
    k_l2_wmma<<<(thr_l2 + B - 1) / B, B, 0, stream>>>(x1, W2, rW2, b2, rb2, dinv, xw2, acc2);

    long long te2 = (long long)N_EDGES * RED;
    k_edge_scatter<RED><<<(unsigned)((te2 + B - 1) / B), B, 0, stream>>>(row, col, ew, dinv,
                                                                          xw2, acc2);

    k_head<<<(N_NODES + 7) / 8, B, 0, stream>>>(acc2, g2, be2, fcW, fcb, pW, pb, logits, emb);
}